// GaussianScene_8022998909760
// MI455X (gfx1250) — compile-verified
//
#include <hip/hip_runtime.h>
#include <cstdint>

// ---------------------------------------------------------------------------
// Gaussian-splat tile rasterizer for gfx1250 (MI455X).
//  grid  = 256 blocks  (one per 16x16 tile of the 256x256 image)
//  block = 256 threads (8 wave32s; wave w owns image rows 2w, 2w+1 of the tile)
// Pipeline per 512-gaussian chunk:
//   async DMA chunk -> LDS (double buffered, ASYNCcnt)        [CDNA5 async path]
//   wave0 ballot-compacts in-tile gaussians into packed LDS SoA [wave32 ballot]
//   all waves composite; color accum via v_wmma_f32_16x16x4_f32 in fp32 [WMMA]
// Cross-half lane exchange for the A fragment uses ds_swizzle SWAPX16.
// ---------------------------------------------------------------------------

#define WIMG 256
#define HIMG 256
#define NG   8192
#define TILE 16
#define CH   512                 // gaussians staged per chunk
#define NCHUNKS (NG / CH)        // 16
#define MINW 1e-6f

// staged chunk layout (bytes): pts 4096 | invcov 8192 | colors 6144 | op 2048 | rad 2048
#define STAGE_BYTES 22528
#define SLOTS 1408               // STAGE_BYTES/16 16B transfers per chunk

typedef float v2f __attribute__((ext_vector_type(2)));
typedef float v4f __attribute__((ext_vector_type(4)));
typedef float v8f __attribute__((ext_vector_type(8)));

__device__ __forceinline__ v8f wmma4(v2f a, v2f b, v8f c) {
  // D(16x16,f32) = A(16x4,f32) x B(4x16,f32) + C   -- full fp32, matches reference
  return __builtin_amdgcn_wmma_f32_16x16x4_f32(false, a, false, b, (short)0, c,
                                               false, false);
}

// exchange value with lane^16 (group-of-32 swizzle, SWAPX16: xor=0x10 and=0x1f)
__device__ __forceinline__ float swap16(float x) {
  return __int_as_float(__builtin_amdgcn_ds_swizzle(__float_as_int(x), 0x401F));
}

__global__ void __launch_bounds__(256)
gsplat_kernel(const float* __restrict__ g_pts,   // [N,2]
              const float* __restrict__ g_ic,    // [N,2,2] (a=i0, b=i1, c=i3)
              const float* __restrict__ g_rad,   // [N]
              const float* __restrict__ g_col,   // [N,3]
              const float* __restrict__ g_op,    // [N] pre-sigmoid
              float* __restrict__ out)           // [W,H,3] flat (x*H+y)*3+c
{
  __shared__ __attribute__((aligned(16))) char stage[2][STAGE_BYTES]; // 44 KB
  __shared__ __attribute__((aligned(16))) v4f cAB[CH];  // {mx, my, ic_a, ic_b}
  __shared__ __attribute__((aligned(16))) v2f cCD[CH];  // {ic_c, sigmoid(op)}
  __shared__ float ccol[CH * 3];
  __shared__ int   s_cnt;

  const int t    = threadIdx.x;
  const int wv   = t >> 5;
  const int lane = t & 31;                  // wave32
  const int tix  = blockIdx.x & 15, tiy = blockIdx.x >> 4;
  const float tx0 = (float)(tix * TILE), ty0 = (float)(tiy * TILE);

  // this lane's pixel: x = tx0 + lane%16, y = ty0 + 2*wv + lane/16
  const int   n   = lane & 15;              // also WMMA column (= color channel)
  const bool  hi  = lane >= 16;
  const float pxf = tx0 + (float)(lane & 15);
  const float pyf = ty0 + (float)(2 * wv + (lane >> 4));

  float T = 1.0f;                           // remaining transmittance
  v8f acc0 = {0,0,0,0,0,0,0,0};             // row y = ty0+2*wv   (M = x_local)
  v8f acc1 = {0,0,0,0,0,0,0,0};             // row y = ty0+2*wv+1

  // -- async DMA of one chunk into a stage buffer: 6 b128 ops per wave ------
  auto stage_issue = [&](int chunk, int b) {
    const char* g0 = (const char*)(g_pts + (size_t)chunk * 1024);
    const char* g1 = (const char*)(g_ic  + (size_t)chunk * 2048);
    const char* g2 = (const char*)(g_col + (size_t)chunk * 1536);
    const char* g3 = (const char*)(g_op  + (size_t)chunk * 512);
    const char* g4 = (const char*)(g_rad + (size_t)chunk * 512);
    uint32_t lbase = (uint32_t)(uintptr_t)&stage[b][0]; // low 32 bits of generic
#pragma unroll
    for (int i = 0; i < 6; ++i) {
      int s = t + 256 * i;
      s = (s < SLOTS - 1) ? s : (SLOTS - 1);            // clamp: uniform op count
      const char* src = (s < 256)  ? g0 + (size_t)s * 16
                      : (s < 768)  ? g1 + (size_t)(s - 256) * 16
                      : (s < 1152) ? g2 + (size_t)(s - 768) * 16
                      : (s < 1280) ? g3 + (size_t)(s - 1152) * 16
                      :              g4 + (size_t)(s - 1280) * 16;
      uint32_t lds = lbase + (uint32_t)(s * 16);
      asm volatile("global_load_async_to_lds_b128 %0, %1, off"
                   :: "v"(lds), "v"(src) : "memory");
    }
  };

  stage_issue(0, 0);

  for (int c = 0; c < NCHUNKS; ++c) {
    if (c + 1 < NCHUNKS) {
      stage_issue(c + 1, (c + 1) & 1);                  // prefetch next chunk
      asm volatile("s_wait_asynccnt 6" ::: "memory");   // current chunk landed
    } else {
      asm volatile("s_wait_asynccnt 0" ::: "memory");
    }
    __syncthreads();                                    // all waves' DMA visible

    const char*  sb  = stage[c & 1];
    const float* pt  = (const float*)(sb + 0);
    const float* ics = (const float*)(sb + 4096);
    const float* cls = (const float*)(sb + 12288);
    const float* ops = (const float*)(sb + 18432);
    const float* rds = (const float*)(sb + 20480);

    // ---- wave0: order-preserving ballot compaction of in-tile gaussians ----
    if (t < 32) {
      int cnt = 0;
      for (int base = 0; base < CH; base += 32) {
        int   i  = base + lane;
        float mx = pt[2 * i], my = pt[2 * i + 1], r = rds[i];
        bool pred = (floorf(mx - r) <= tx0 + 16.0f) && (ceilf(mx + r) >= tx0)
                 && (floorf(my - r) <= ty0 + 16.0f) && (ceilf(my + r) >= ty0);
        unsigned long long m = __ballot(pred ? 1 : 0);
        int pos = cnt + (int)__popcll(m & ((1ull << lane) - 1ull));
        if (pred) {
          v4f ab; ab.x = mx; ab.y = my;
          ab.z = ics[4 * i + 0]; ab.w = ics[4 * i + 1];
          cAB[pos] = ab;
          v2f cd; cd.x = ics[4 * i + 3];
          cd.y = 1.0f / (1.0f + __expf(-ops[i]));       // sigmoid once/gaussian
          cCD[pos] = cd;
          ccol[3 * pos + 0] = cls[3 * i + 0];
          ccol[3 * pos + 1] = cls[3 * i + 1];
          ccol[3 * pos + 2] = cls[3 * i + 2];
        }
        cnt += (int)__popcll(m);
      }
      int cnt4 = (cnt + 3) & ~3;                        // pad to WMMA K-group
      if (lane < cnt4 - cnt) {
        int p = cnt + lane;                             // zero-opacity padding
        v4f z4 = {0,0,0,0};  v2f z2 = {0,0};
        cAB[p] = z4;  cCD[p] = z2;
        ccol[3 * p] = ccol[3 * p + 1] = ccol[3 * p + 2] = 0.0f;
      }
      if (lane == 0) s_cnt = cnt4;
    }
    __syncthreads();

    const int cnt4 = s_cnt;                             // workgroup-uniform

    // ---- composite (wave-uniform control flow; EXEC all-ones at WMMA) ------
    if (!__all(T < MINW)) {
      auto gstep = [&](int i) -> float {
        v4f ab = cAB[i];                                // one ds_load_b128
        v2f cd = cCD[i];                                // one ds_load_b64
        float dx = pxf - ab.x, dy = pyf - ab.y;
        float q  = ab.z * dx * dx + 2.0f * ab.w * dx * dy + cd.x * dy * dy;
        float alpha = __expf(-0.5f * q) * cd.y;
        float Tn = T * (1.0f - alpha);
        float contrib = (Tn >= MINW) ? T * alpha : 0.0f; // include iff T_after>=1e-6
        T = Tn;
        return contrib;
      };

      const int gb0o = hi ? 2 : 0, gb1o = hi ? 3 : 1;
      const int nc   = (n < 3) ? n : 0;
      const float colmask = (n < 3) ? 1.0f : 0.0f;

      for (int g = 0; g < cnt4; g += 4) {
        float cc0 = gstep(g + 0);
        float cc1 = gstep(g + 1);
        float cc2 = gstep(g + 2);
        float cc3 = gstep(g + 3);

        // A fragments (16x4 f32): lanes 0-15 hold K=0,1; lanes 16-31 hold K=2,3
        float x02 = swap16(cc2);           // lane l <- lane l^16's cc2
        float x13 = swap16(cc3);
        float x20 = swap16(cc0);
        float x31 = swap16(cc1);
        v2f A0, A1, B;
        A0.x = hi ? x02 : cc0;  A0.y = hi ? x13 : cc1;   // pixels 0..15 of wave
        A1.x = hi ? cc2 : x20;  A1.y = hi ? cc3 : x31;   // pixels 16..31 of wave

        // B fragment (4x16 f32): VGPR0 rows K=0|2, VGPR1 rows K=1|3; cols=RGB
        B.x = ccol[3 * (g + gb0o) + nc] * colmask;
        B.y = ccol[3 * (g + gb1o) + nc] * colmask;

        acc0 = wmma4(A0, B, acc0);
        acc1 = wmma4(A1, B, acc1);

        if (__all(T < MINW)) break;                      // wave-uniform early out
      }
    }
    __syncthreads();                                     // before stage reuse
  }

  // ---- writeback: D layout => lane holds column n, rows j(+8 if hi) --------
  if (n < 3) {
    const int ybase = tiy * TILE + 2 * wv;
    const int xbase = tix * TILE + (hi ? 8 : 0);
#pragma unroll
    for (int j = 0; j < 8; ++j) {
      int x = xbase + j;
      out[((size_t)x * HIMG + ybase) * 3 + n]       = acc0[j];
      out[((size_t)x * HIMG + (ybase + 1)) * 3 + n] = acc1[j];
    }
  }
}

extern "C" void kernel_launch(void* const* d_in, const int* in_sizes, int n_in,
                              void* d_out, int out_size, void* d_ws, size_t ws_size,
                              hipStream_t stream) {
  (void)in_sizes; (void)n_in; (void)d_ws; (void)ws_size; (void)out_size;
  const float* pts = (const float*)d_in[0];   // points_xy        [8192,2]
  const float* ic  = (const float*)d_in[1];   // inverse_cov      [8192,2,2]
  const float* rad = (const float*)d_in[2];   // radius           [8192]
  const float* col = (const float*)d_in[3];   // colors           [8192,3]
  const float* op  = (const float*)d_in[4];   // opacity logits   [8192]
  float* out = (float*)d_out;                 // [256,256,3]
  gsplat_kernel<<<dim3(256), dim3(256), 0, stream>>>(pts, ic, rad, col, op, out);
}